// Model_40183714021719
// MI455X (gfx1250) — compile-verified
//
#include <hip/hip_runtime.h>
#include <math.h>

#define EPSN 1e-8f
#define RAD2 (0.03f * 0.03f)

typedef __attribute__((ext_vector_type(2))) float v2f;
typedef __attribute__((ext_vector_type(4))) float v4f;
typedef __attribute__((ext_vector_type(8))) float v8f;

// ---------------- precompute world mean / inv(max(std,eps)) into ws ----------------
__global__ void world_params_kernel(const float* __restrict__ wmean,
                                    const float* __restrict__ wstd,
                                    float* __restrict__ params /* 8 floats */)
{
    int i = threadIdx.x;
    if (i < 4) {
        params[i]     = wmean[i];
        params[4 + i] = 1.0f / fmaxf(wstd[i], EPSN);   // precise div, once per dispatch
    }
}

// ---------------- node features: [N,12] = (vel | one_hot) normalized ----------------
__global__ void node_feat_kernel(const float* __restrict__ wp,
                                 const float* __restrict__ pwp,
                                 const int*   __restrict__ ntype,
                                 const float* __restrict__ nmean,
                                 const float* __restrict__ nstd,
                                 float* __restrict__ nout, int n)
{
    int i = blockIdx.x * blockDim.x + threadIdx.x;
    if (i >= n) return;
    float vel[3];
    vel[0] = wp[3*i+0] - pwp[3*i+0];
    vel[1] = wp[3*i+1] - pwp[3*i+1];
    vel[2] = wp[3*i+2] - pwp[3*i+2];
    int t = ntype[i];
#pragma unroll
    for (int k = 0; k < 12; ++k) {
        float f = (k < 3) ? vel[k] : ((k - 3 == t) ? 1.0f : 0.0f);
        nout[i*12 + k] = (f - nmean[k]) / fmaxf(nstd[k], EPSN);
    }
}

// ---------------- zero agg region (harness poisons d_out) ----------------
__global__ void zero_kernel(float* __restrict__ p, int count)
{
    int i = blockIdx.x * blockDim.x + threadIdx.x;
    if (i < count) p[i] = 0.0f;
}

// ---------------- mesh edge features + segment_sum scatter ----------------
__global__ void mesh_edge_kernel(const float* __restrict__ wp,
                                 const float* __restrict__ mp,
                                 const int*   __restrict__ snd,
                                 const int*   __restrict__ rcv,
                                 const float* __restrict__ mmean,
                                 const float* __restrict__ mstd,
                                 float* __restrict__ mout,
                                 float* __restrict__ agg, int E)
{
    int e = blockIdx.x * blockDim.x + threadIdx.x;
    if (e >= E) return;
    int s = snd[e], r = rcv[e];
    float rwx = wp[3*s+0] - wp[3*r+0];
    float rwy = wp[3*s+1] - wp[3*r+1];
    float rwz = wp[3*s+2] - wp[3*r+2];
    float sw  = rwx*rwx + rwy*rwy + rwz*rwz;
    float nw  = (sw > 0.0f) ? sqrtf(sw) : 0.0f;
    float rmx = mp[2*s+0] - mp[2*r+0];
    float rmy = mp[2*s+1] - mp[2*r+1];
    float sm  = rmx*rmx + rmy*rmy;
    float nm  = (sm > 0.0f) ? sqrtf(sm) : 0.0f;
    float f[7] = { rwx, rwy, rwz, nw, rmx, rmy, nm };
#pragma unroll
    for (int k = 0; k < 7; ++k) {
        float o = (f[k] - mmean[k]) / fmaxf(mstd[k], EPSN);
        mout[(size_t)e*7 + k] = o;
        atomicAdd(&agg[(size_t)r*7 + k], o);
    }
}

// ---------------- world tiles: one wave32 = one 16x16 (i,j) tile via WMMA ----------------
__global__ void world_tile_kernel(const float* __restrict__ wp,
                                  const int*   __restrict__ ntype,
                                  const float* __restrict__ params, /* m0..m3, is0..is3 */
                                  float* __restrict__ wout,
                                  int n, int tpr, int ntiles)
{
    int gtid = blockIdx.x * blockDim.x + threadIdx.x;
    int wave = gtid >> 5;
    int lane = threadIdx.x & 31;
    if (wave >= ntiles) return;        // wave-uniform: EXEC stays all-ones below

    int ti = wave / tpr;
    int tj = wave - ti * tpr;
    int i0 = ti << 4, j0 = tj << 4;
    int half = lane >> 4;              // K-group selector per ISA 16x4 f32 layout
    int l    = lane & 15;

    // B-matrix column (this lane's j) + per-column scalars
    int colr = j0 + l;
    int col  = (colr < n) ? colr : (n - 1);
    float pjx = wp[3*col+0], pjy = wp[3*col+1], pjz = wp[3*col+2];
    float sqj = pjx*pjx + pjy*pjy + pjz*pjz;
    bool  obst = (ntype[col] == 1);

    // A-matrix row for this lane (lanes 0-15: K=0,1 ; lanes 16-31: K=2,3)
    int arowr = i0 + l;
    int arow  = (arowr < n) ? arowr : (n - 1);
    v2f a, b;
    a.x = wp[3*arow + (half ? 2 : 0)];
    a.y = half ? 0.0f : wp[3*arow + 1];
    b.x = half ? pjz : pjx;
    b.y = half ? 0.0f : pjy;

    v8f c = {};
    // D[M][N] = pos_i . pos_j  for the whole 16x16 tile in one matrix op
    v8f d = __builtin_amdgcn_wmma_f32_16x16x4_f32(false, a, false, b,
                                                  (short)0, c, false, false);

    // loop-invariant normalization constants, precomputed once per dispatch
    float m0  = params[0], m1  = params[1], m2  = params[2], m3  = params[3];
    float is0 = params[4], is1 = params[5], is2 = params[6], is3 = params[7];

    int rbase = i0 + (half << 3);      // VGPR v of this lane holds M = rbase + v, N = l
    bool colok = (colr < n);
#pragma unroll
    for (int v = 0; v < 8; ++v) {
        int rowr = rbase + v;
        int row  = (rowr < n) ? rowr : (n - 1);
        float pix = wp[3*row+0], piy = wp[3*row+1], piz = wp[3*row+2];
        float sqi = pix*pix + piy*piy + piz*piz;
        float d2  = sqi + sqj - 2.0f * d[v];          // expansion form, matches reference
        float rx = pjx - pix, ry = pjy - piy, rz = pjz - piz;
        float s2 = rx*rx + ry*ry + rz*rz;
        float nrm = (s2 > 0.0f) ? sqrtf(s2) : 0.0f;   // _safe_norm
        bool conn = (d2 < RAD2) && (row != col) && (!obst);
        v4f o;
        o.x = conn ? (rx  - m0) * is0 : 0.0f;
        o.y = conn ? (ry  - m1) * is1 : 0.0f;
        o.z = conn ? (rz  - m2) * is2 : 0.0f;
        o.w = conn ? (nrm - m3) * is3 : 0.0f;
        if (colok && (rowr < n)) {
            // 576 MB total >> 192 MB L2: stream past L2 with non-temporal store
            __builtin_nontemporal_store(o, (v4f*)(wout + ((size_t)row * n + col) * 4));
        }
    }
}

// ---------------- knock out existing mesh edges: conn.at[s,r] = False ----------------
__global__ void edge_clear_kernel(const int* __restrict__ snd,
                                  const int* __restrict__ rcv,
                                  float* __restrict__ wout, int n, int E)
{
    int e = blockIdx.x * blockDim.x + threadIdx.x;
    if (e >= E) return;
    int s = snd[e], r = rcv[e];
    v4f z = {};
    *(v4f*)(wout + ((size_t)s * n + r) * 4) = z;
}

extern "C" void kernel_launch(void* const* d_in, const int* in_sizes, int n_in,
                              void* d_out, int out_size, void* d_ws, size_t ws_size,
                              hipStream_t stream)
{
    const float* wp    = (const float*)d_in[0];
    const float* pwp   = (const float*)d_in[1];
    const float* mp    = (const float*)d_in[2];
    const int*   ntype = (const int*)  d_in[3];
    const int*   snd   = (const int*)  d_in[4];
    const int*   rcv   = (const int*)  d_in[5];
    const float* nmean = (const float*)d_in[6];
    const float* nstd  = (const float*)d_in[7];
    const float* mmean = (const float*)d_in[8];
    const float* mstd  = (const float*)d_in[9];
    const float* wmean = (const float*)d_in[10];
    const float* wstd  = (const float*)d_in[11];

    int n = in_sizes[0] / 3;   // 6000
    int E = in_sizes[4];       // 48000

    float* out       = (float*)d_out;
    float* node_out  = out;
    float* mesh_out  = node_out + (size_t)n * 12;
    float* world_out = mesh_out + (size_t)E * 7;
    float* agg       = world_out + (size_t)n * n * 4;

    float* params = (float*)d_ws;   // 8 floats of scratch

    world_params_kernel<<<1, 32, 0, stream>>>(wmean, wstd, params);

    node_feat_kernel<<<(n + 255) / 256, 256, 0, stream>>>(wp, pwp, ntype, nmean, nstd, node_out, n);

    int aggN = n * 7;
    zero_kernel<<<(aggN + 255) / 256, 256, 0, stream>>>(agg, aggN);

    mesh_edge_kernel<<<(E + 255) / 256, 256, 0, stream>>>(wp, mp, snd, rcv, mmean, mstd,
                                                          mesh_out, agg, E);

    int tpr = (n + 15) / 16;
    int ntiles = tpr * tpr;                 // 375*375 = 140625 tiles
    int wavesPerBlock = 256 / 32;           // wave32: 8 waves per 256-thread block
    int blocks = (ntiles + wavesPerBlock - 1) / wavesPerBlock;
    world_tile_kernel<<<blocks, 256, 0, stream>>>(wp, ntype, params, world_out, n, tpr, ntiles);

    edge_clear_kernel<<<(E + 255) / 256, 256, 0, stream>>>(snd, rcv, world_out, n, E);
}